// Yolo_postprocessor_33818572489443
// MI455X (gfx1250) — compile-verified
//
#include <hip/hip_runtime.h>
#include <math.h>

typedef __attribute__((ext_vector_type(2))) float v2f;
typedef __attribute__((ext_vector_type(8))) float v8f;

#define N_DET 8192
#define NW    (N_DET / 32)   // 256 mask words per row
#define NT    (N_DET / 16)   // 512 i-tiles (16 rows each)
#define NS    (N_DET / 32)   // 256 j-strips (32 cols each)
#define CONF_T 0.25f
#define IOU_T  0.45f

// ---------------------------------------------------------------------------
// K1: decode each detection -> clamped xyxy box, sortable key, valid flag
// ---------------------------------------------------------------------------
__global__ void k_decode(const float* __restrict__ x, int imgW, int imgH,
                         float4* __restrict__ xyxy, unsigned* __restrict__ key32,
                         unsigned* __restrict__ validArr) {
  int i = blockIdx.x * blockDim.x + threadIdx.x;
  if (i >= N_DET) return;
  const float* r = x + (size_t)i * 85;
  float conf = r[4];
  float c0   = r[5];
  float mx   = r[6];
  #pragma unroll 4
  for (int c = 7; c < 85; ++c) mx = fmaxf(mx, r[c]);
  bool valid = (conf > CONF_T) && (c0 >= mx);        // argmax over classes == 0
  float b0 = r[0] * (float)imgH, b1 = r[1] * (float)imgW;
  float b2 = r[2] * (float)imgH, b3 = r[3] * (float)imgW;
  valid = valid && ((b2 > 0.0f) || (b3 > 0.0f));
  float hw = truncf(b2 * 0.5f), hh = truncf(b3 * 0.5f);
  int x1 = (int)(b0 - hw), y1 = (int)(b1 - hh);
  int x2 = (int)(b0 + hw), y2 = (int)(b1 + hh);
  x1 = max(x1, 0); y1 = max(y1, 0); x2 = min(x2, imgW); y2 = min(y2, imgH);
  xyxy[i] = make_float4((float)x1, (float)y1, (float)x2, (float)y2);
  // descending-score key with ascending-index tiebreak (matches stable argsort)
  float score = valid ? conf : -1.0f;
  unsigned u = __float_as_uint(score);
  u = ((int)u < 0) ? ~u : (u | 0x80000000u);   // monotonic ascending transform
  key32[i] = ~u;                                // invert -> ascending == desc score
  validArr[i] = valid ? 1u : 0u;
}

// ---------------------------------------------------------------------------
// K2: single-workgroup bitonic sort of 8192 (key<<32 | idx) in 64 KB LDS
// ---------------------------------------------------------------------------
__global__ void __launch_bounds__(1024)
k_sort(const unsigned* __restrict__ key32, const unsigned* __restrict__ validArr,
       unsigned* __restrict__ order, unsigned* __restrict__ validmask) {
  __shared__ unsigned long long sk[N_DET];   // 64 KB
  int tid = threadIdx.x;
  for (int t = tid; t < N_DET; t += 1024)
    sk[t] = ((unsigned long long)key32[t] << 32) | (unsigned)t;
  __syncthreads();
  for (int k = 2; k <= N_DET; k <<= 1) {
    for (int j = k >> 1; j > 0; j >>= 1) {
      for (int t = tid; t < N_DET; t += 1024) {
        int p = t ^ j;
        if (p > t) {
          bool up = ((t & k) == 0);
          unsigned long long a = sk[t], b = sk[p];
          if ((a > b) == up) { sk[t] = b; sk[p] = a; }
        }
      }
      __syncthreads();
    }
  }
  for (int t = tid; t < N_DET; t += 1024)
    order[t] = (unsigned)(sk[t] & 0xffffffffu);
  for (int w = tid; w < NW; w += 1024) {
    unsigned m = 0;
    #pragma unroll 4
    for (int b = 0; b < 32; ++b) {
      unsigned idx = (unsigned)(sk[w * 32 + b] & 0xffffffffu);
      m |= (validArr[idx] & 1u) << b;
    }
    validmask[w] = m;
  }
}

// ---------------------------------------------------------------------------
// K3: gather sorted boxes (packed float4 -> b128 traffic) + precompute areas
// ---------------------------------------------------------------------------
__global__ void k_gather(const unsigned* __restrict__ order,
                         const float4* __restrict__ xyxy,
                         float4* __restrict__ boxs, float* __restrict__ areas) {
  int k = blockIdx.x * blockDim.x + threadIdx.x;
  if (k >= N_DET) return;
  float4 b = xyxy[order[k]];
  boxs[k] = b;
  areas[k] = (b.z - b.x) * (b.w - b.y);
}

// ---------------------------------------------------------------------------
// K4: suppression bitmask. One wave per (16-row i-tile x 32-col j-strip).
// union term area_i + area_j via V_WMMA_F32_16X16X4_F32 outer-sum:
//   A[m,:] = [area_i[m], 1, 0, 0] ;  B[:,n] = [1, area_j[n], 0, 0]^T
// C/D f32 layout: VGPR v, lanes 0-15 -> M=v, lanes 16-31 -> M=v+8; N = lane&15.
// Boxes loaded as float4 (global_load_b128) to minimize VMEM issue count.
// ---------------------------------------------------------------------------
__global__ void __launch_bounds__(256)
k_suppmat(const float4* __restrict__ boxs, const float* __restrict__ areas,
          unsigned* __restrict__ supp) {
  int lane = threadIdx.x & 31;
  int wid  = blockIdx.x * 8 + (threadIdx.x >> 5);
  int ti = wid >> 8;          // 0..511
  int tj = wid & 255;         // 0..255 (32-wide j strip == one mask word)
  int i0 = ti * 16;
  int n  = lane & 15;
  int mh = lane >> 4;

  int jA = tj * 32 + n;        // j half 0
  int jB = tj * 32 + 16 + n;   // j half 1
  float4 bja = boxs[jA];       // one b128 each
  float4 bjb = boxs[jB];
  float arA = areas[jA];
  float arB = areas[jB];
  float ai  = areas[i0 + n];   // in-range for all lanes

  v2f a, bA, bB;
  a.x  = mh ? 0.0f : ai;   a.y  = mh ? 0.0f : 1.0f;   // K=0: area_i, K=1: 1
  bA.x = mh ? 0.0f : 1.0f; bA.y = mh ? 0.0f : arA;    // K=0: 1, K=1: area_j
  bB.x = mh ? 0.0f : 1.0f; bB.y = mh ? 0.0f : arB;
  v8f zero = {0.f, 0.f, 0.f, 0.f, 0.f, 0.f, 0.f, 0.f};

  v8f d0 = __builtin_amdgcn_wmma_f32_16x16x4_f32(false, a, false, bA,
                                                 (short)0, zero, false, false);
  v8f d1 = __builtin_amdgcn_wmma_f32_16x16x4_f32(false, a, false, bB,
                                                 (short)0, zero, false, false);

  unsigned myword = 0;
  #pragma unroll
  for (int v = 0; v < 8; ++v) {
    int ig = i0 + v + mh * 8;                  // global row in sorted order
    float4 bi = boxs[ig];                      // broadcast b128 load
    // half A
    float iwA = fmaxf(fminf(bi.z, bja.z) - fmaxf(bi.x, bja.x), 0.0f);
    float ihA = fmaxf(fminf(bi.w, bja.w) - fmaxf(bi.y, bja.y), 0.0f);
    float inA = iwA * ihA;
    float uA  = fmaxf(d0[v] - inA, 1e-9f);     // union = area_i+area_j - inter
    bool cA = (inA > IOU_T * uA) && (jA > ig);
    // half B
    float iwB = fmaxf(fminf(bi.z, bjb.z) - fmaxf(bi.x, bjb.x), 0.0f);
    float ihB = fmaxf(fminf(bi.w, bjb.w) - fmaxf(bi.y, bjb.y), 0.0f);
    float inB = iwB * ihB;
    float uB  = fmaxf(d1[v] - inB, 1e-9f);
    bool cB = (inB > IOU_T * uB) && (jB > ig);

    unsigned b0 = __builtin_amdgcn_ballot_w32(cA);  // lo16: M=v, hi16: M=v+8
    unsigned b1 = __builtin_amdgcn_ballot_w32(cB);
    unsigned lowrow = (b0 & 0xffffu) | (b1 << 16);
    unsigned hirow  = (b0 >> 16)     | (b1 & 0xffff0000u);
    if (lane == v)     myword = lowrow;
    if (lane == v + 8) myword = hirow;
  }
  if (lane < 16)
    supp[(size_t)(i0 + lane) * NW + tj] = myword;   // full coverage, no clear
}

// ---------------------------------------------------------------------------
// K5: sequential NMS scan on ONE wave32. alive mask (8192 bits) lives in
// 8 VGPRs/lane; word w owned by lane (w&31), slot (w>>5). Decision bit is
// broadcast with ds_bpermute; kept rows clear their suppression word-row from alive.
// ---------------------------------------------------------------------------
__global__ void __launch_bounds__(32)
k_scan(const unsigned* __restrict__ validmask, const unsigned* __restrict__ supp,
       unsigned* __restrict__ aliveOut) {
  int lane = threadIdx.x;
  unsigned alive[8];
  #pragma unroll
  for (int k = 0; k < 8; ++k) alive[k] = validmask[lane + 32 * k];

  #pragma unroll
  for (int s = 0; s < 8; ++s) {            // slot: i in [s*1024, s*1024+1024)
    for (int ii = 0; ii < 1024; ++ii) {
      int i = s * 1024 + ii;
      int owner = (i >> 5) & 31;           // lane owning word i>>5 (slot==s)
      unsigned aw = (unsigned)__builtin_amdgcn_ds_bpermute(owner << 2,
                                                           (int)alive[s]);
      if ((aw >> (i & 31)) & 1u) {         // wave-uniform
        const unsigned* row = supp + (size_t)i * NW;
        #pragma unroll
        for (int k = 0; k < 8; ++k) alive[k] &= ~row[lane + 32 * k];
      }
    }
  }
  #pragma unroll
  for (int k = 0; k < 8; ++k) aliveOut[lane + 32 * k] = alive[k];
}

// ---------------------------------------------------------------------------
// K6: final gather: boxes_s * keep (8192x4, b128 store) then scores_s * keep
// ---------------------------------------------------------------------------
__global__ void k_output(const unsigned* __restrict__ order,
                         const unsigned* __restrict__ alive,
                         const float4* __restrict__ boxs,
                         const float* __restrict__ x, float* __restrict__ out) {
  int k = blockIdx.x * blockDim.x + threadIdx.x;
  if (k >= N_DET) return;
  float kf = (float)((alive[k >> 5] >> (k & 31)) & 1u);
  float4 b = boxs[k];
  ((float4*)out)[k] = make_float4(b.x * kf, b.y * kf, b.z * kf, b.w * kf);
  unsigned j = order[k];
  out[N_DET * 4 + k] = x[(size_t)j * 85 + 4] * kf;   // raw conf, gathered
}

// ---------------------------------------------------------------------------
extern "C" void kernel_launch(void* const* d_in, const int* in_sizes, int n_in,
                              void* d_out, int out_size, void* d_ws, size_t ws_size,
                              hipStream_t stream) {
  const float* x = (const float*)d_in[0];
  int dim = (int)lround(sqrt((double)in_sizes[1] / 3.0));   // 640
  int imgW = dim, imgH = dim;

  char* ws = (char*)d_ws;
  size_t off = 0;
  float4*   xyxy      = (float4*)(ws + off);   off += (size_t)N_DET * 16;
  float4*   boxs      = (float4*)(ws + off);   off += (size_t)N_DET * 16;
  unsigned* key32     = (unsigned*)(ws + off); off += (size_t)N_DET * 4;
  unsigned* validArr  = (unsigned*)(ws + off); off += (size_t)N_DET * 4;
  unsigned* order     = (unsigned*)(ws + off); off += (size_t)N_DET * 4;
  float*    areas     = (float*)(ws + off);    off += (size_t)N_DET * 4;
  unsigned* validmask = (unsigned*)(ws + off); off += (size_t)NW * 4;
  unsigned* aliveOut  = (unsigned*)(ws + off); off += (size_t)NW * 4;
  unsigned* supp      = (unsigned*)(ws + off); off += (size_t)N_DET * NW * 4; // 8 MB

  k_decode<<<N_DET / 256, 256, 0, stream>>>(x, imgW, imgH, xyxy, key32, validArr);
  k_sort<<<1, 1024, 0, stream>>>(key32, validArr, order, validmask);
  k_gather<<<N_DET / 256, 256, 0, stream>>>(order, xyxy, boxs, areas);
  k_suppmat<<<(NT * NS) / 8, 256, 0, stream>>>(boxs, areas, supp);
  k_scan<<<1, 32, 0, stream>>>(validmask, supp, aliveOut);
  k_output<<<N_DET / 256, 256, 0, stream>>>(order, aliveOut, boxs, x, (float*)d_out);
}